// MessagePassing_26345329394134
// MI455X (gfx1250) — compile-verified
//
#include <hip/hip_runtime.h>
#include <cstdint>
#include <cstddef>

// ---------------------------------------------------------------------------
// Equivariant message passing (e3nn-style tensor product + scatter-add).
// Memory-bound (AI ~0.25 flop/byte). Optimizations:
//  * weights (320MB, read exactly once) streamed via CDNA5 async load-to-LDS
//    (global_load_async_to_lds_b128/b64, th:TH_LOAD_NT), double-buffered per
//    wave, released with s_wait_asynccnt — deep prefetch, zero VGPR cost.
//  * wave-uniform edge metadata (senders/receivers/sh) forced onto the scalar
//    path (s_load) via readfirstlane.
//  * node gathers issued BEFORE the asynccnt wait so vector loads overlap it.
//  * scatter via non-returning agent-scope global_atomic_add_f32.
// No WMMA: the op has no K>=16 contraction; matrix ceilings are irrelevant
// at this arithmetic intensity.
// ---------------------------------------------------------------------------

#define WPB      8        // waves per block (wave32 -> 256 threads)
#define NBLOCKS  2048
#define WROW     320      // floats per edge weight row (5 paths * 64 mul)

__device__ __forceinline__ uint32_t lds_off(const void* p) {
  // Low 32 bits of the flat LDS address == wave-relative LDS byte offset.
  return (uint32_t)(uintptr_t)p;
}

// Stage one 1280-byte weight row into LDS via the async engine.
// 2x b128 (32 lanes * 16B) + 1x b64 (32 lanes * 8B). INST_OFFSET is added to
// both the global and the LDS address.
__device__ __forceinline__ void async_copy_wrow(const float* wrow,
                                                uint32_t ldsBase, int lane) {
  const uint64_t gp   = (uint64_t)(uintptr_t)wrow;
  const uint32_t off  = (uint32_t)lane * 16u;
  const uint32_t lds  = ldsBase + off;
  asm volatile("global_load_async_to_lds_b128 %0, %1, %2 offset:0 th:TH_LOAD_NT"
               :: "v"(lds), "v"(off), "s"(gp) : "memory");
  asm volatile("global_load_async_to_lds_b128 %0, %1, %2 offset:512 th:TH_LOAD_NT"
               :: "v"(lds), "v"(off), "s"(gp) : "memory");
  const uint32_t off2 = (uint32_t)lane * 8u;
  const uint32_t lds2 = ldsBase + off2;
  asm volatile("global_load_async_to_lds_b64 %0, %1, %2 offset:1024 th:TH_LOAD_NT"
               :: "v"(lds2), "v"(off2), "s"(gp) : "memory");
}

__device__ __forceinline__ void atomAddF32(float* p, float v) {
  // Relaxed agent-scope fp32 add -> non-returning global_atomic_add_f32.
  __hip_atomic_fetch_add(p, v, __ATOMIC_RELAXED, __HIP_MEMORY_SCOPE_AGENT);
}

__global__ void __launch_bounds__(WPB * 32)
mp_edge_kernel(const float* __restrict__ weights,
               const float* __restrict__ x0,
               const float* __restrict__ x1,
               const float* __restrict__ sh0,
               const float* __restrict__ sh1,
               const int*   __restrict__ senders,
               const int*   __restrict__ receivers,
               float*       __restrict__ out,   // out0 [N*128] then out1 [N*576]
               int E, int N)
{
  constexpr float EPS        = 0.125f;
  constexpr float INV_SQRT3  = 0.57735026918962576451f;
  constexpr float INV_SQRT2  = 0.70710678118654752440f;

  __shared__ float smem[WPB][2][WROW];   // double buffer per wave (20 KB/block)

  const int lane = (int)(threadIdx.x & 31u);
  // Force wave-uniformity so edge metadata loads lower to s_load.
  const int wid  = __builtin_amdgcn_readfirstlane((int)(threadIdx.x >> 5u));
  const int gw     = (int)blockIdx.x * WPB + wid;
  const int stride = (int)gridDim.x * WPB;

  float* const out1base = out + (size_t)N * 128;

  const uint32_t ldsA = lds_off(&smem[wid][0][0]);
  const uint32_t ldsB = lds_off(&smem[wid][1][0]);

  int e = gw;
  if (e < E) async_copy_wrow(weights + (size_t)e * WROW, ldsA, lane);

  int buf = 0;
  for (; e < E; e += stride) {
    const int  en = e + stride;
    const bool pf = (en < E);

    // 1) Prefetch next edge's weights into the other LDS buffer.
    if (pf) async_copy_wrow(weights + (size_t)en * WROW, buf ? ldsA : ldsB, lane);

    // 2) Issue this edge's gathers (scalar metadata + vector node data) BEFORE
    //    waiting on the async copy, so they overlap the wait.
    const int   s   = senders[e];          // uniform -> s_load
    const int   r   = receivers[e];        // uniform -> s_load
    const float s0  = sh0[e];              // uniform -> s_load
    const float s10 = sh1[3 * e + 0];
    const float s11 = sh1[3 * e + 1];
    const float s12 = sh1[3 * e + 2];

    const float* __restrict__ x0row = x0 + (size_t)s * 64;
    const float* __restrict__ x1row = x1 + (size_t)s * 192;

    float a[2], b0[2], b1[2], b2[2];
#pragma unroll
    for (int h = 0; h < 2; ++h) {
      const int m = lane + h * 32;
      a[h]  = x0row[m];
      b0[h] = x1row[3 * m + 0];
      b1[h] = x1row[3 * m + 1];
      b2[h] = x1row[3 * m + 2];
    }

    // 3) Release the current buffer's async triple (in-order completion:
    //    asynccnt<=3 after issuing 3 new ops means the older 3 are done).
    if (pf) asm volatile("s_wait_asynccnt 3" ::: "memory");
    else    asm volatile("s_wait_asynccnt 0" ::: "memory");

    const float* __restrict__ w = &smem[wid][buf][0];
    float* const o0 = out      + (size_t)r * 128;
    float* const o1 = out1base + (size_t)r * 576;

#pragma unroll
    for (int h = 0; h < 2; ++h) {
      const int m = lane + h * 32;

      const float w0 = EPS * w[0 * 64 + m];
      const float w1 = EPS * w[1 * 64 + m];
      const float w2 = EPS * w[2 * 64 + m];
      const float w3 = EPS * w[3 * 64 + m];
      const float w4 = EPS * w[4 * 64 + m];

      // msg0: [m00 (c=0..63) | m110 (c=64..127)]
      const float m00  = w0 * a[h] * s0;
      const float dot  = (b0[h] * s10 + b1[h] * s11 + b2[h] * s12) * INV_SQRT3;
      const float m110 = w3 * dot;
      atomAddF32(o0 + m,      m00);
      atomAddF32(o0 + 64 + m, m110);

      // msg1: [m01 | m10 | m111], each (64,3)
      const float m01s = w1 * a[h];
      atomAddF32(o1 + 3 * m + 0, m01s * s10);
      atomAddF32(o1 + 3 * m + 1, m01s * s11);
      atomAddF32(o1 + 3 * m + 2, m01s * s12);

      const float w2s0 = w2 * s0;
      atomAddF32(o1 + 3 * (64 + m) + 0, w2s0 * b0[h]);
      atomAddF32(o1 + 3 * (64 + m) + 1, w2s0 * b1[h]);
      atomAddF32(o1 + 3 * (64 + m) + 2, w2s0 * b2[h]);

      const float c0 = b1[h] * s12 - b2[h] * s11;
      const float c1 = b2[h] * s10 - b0[h] * s12;
      const float c2 = b0[h] * s11 - b1[h] * s10;
      const float w4c = w4 * INV_SQRT2;
      atomAddF32(o1 + 3 * (128 + m) + 0, w4c * c0);
      atomAddF32(o1 + 3 * (128 + m) + 1, w4c * c1);
      atomAddF32(o1 + 3 * (128 + m) + 2, w4c * c2);
    }
    buf ^= 1;
  }
}

__global__ void zero_kernel(float* __restrict__ out, int n) {
  const int i = ((int)blockIdx.x * (int)blockDim.x + (int)threadIdx.x) * 4;
  if (i + 3 < n) {
    *reinterpret_cast<float4*>(out + i) = make_float4(0.f, 0.f, 0.f, 0.f);
  } else {
    for (int k = i; k < n; ++k) out[k] = 0.f;
  }
}

extern "C" void kernel_launch(void* const* d_in, const int* in_sizes, int n_in,
                              void* d_out, int out_size, void* d_ws, size_t ws_size,
                              hipStream_t stream) {
  const float* weights   = (const float*)d_in[0];
  const float* x0        = (const float*)d_in[1];
  const float* x1        = (const float*)d_in[2];
  const float* sh0       = (const float*)d_in[3];
  const float* sh1       = (const float*)d_in[4];
  const int*   senders   = (const int*)d_in[5];
  const int*   receivers = (const int*)d_in[6];

  const int E = in_sizes[0] / WROW;  // 262144
  const int N = in_sizes[1] / 64;    // 65536

  float* out = (float*)d_out;

  // Harness poisons d_out; the scatter needs zeros.
  const int zthreads = (out_size + 3) / 4;
  zero_kernel<<<(zthreads + 255) / 256, 256, 0, stream>>>(out, out_size);

  mp_edge_kernel<<<NBLOCKS, WPB * 32, 0, stream>>>(
      weights, x0, x1, sh0, sh1, senders, receivers, out, E, N);
}